// SingleStageDetector_57105885168323
// MI455X (gfx1250) — compile-verified
//
#include <hip/hip_runtime.h>
#include <hip/hip_bf16.h>
#include <stdint.h>

// Shapes from the reference
#define B_  16
#define A_  9
#define H_  56
#define W_  56
#define N_  64
#define P_  (A_ * H_ * W_)   // 28224 proposals per batch
#define TILE_P 64            // proposals per block

typedef __attribute__((ext_vector_type(2))) float v2f;
typedef __attribute__((ext_vector_type(8))) float v8f;

// gfx1250 async global->LDS copy, 16 bytes per active lane (ASYNCcnt-tracked).
__device__ __forceinline__ void async_copy_b128(const void* gsrc, void* ldst) {
    unsigned lds_addr = (unsigned)(uintptr_t)ldst;   // low 32 bits = LDS byte address
    asm volatile("global_load_async_to_lds_b128 %0, %1, off"
                 :: "v"(lds_addr), "v"(gsrc) : "memory");
}

__device__ __forceinline__ void wait_async0() {
    asm volatile("s_wait_asynccnt 0x0" ::: "memory");
}

__global__ __launch_bounds__(128) void iou_fused_kernel(
    const float* __restrict__ anc,      // (A,2)
    const float* __restrict__ grid,     // (B,H,W,2)
    const float* __restrict__ offsets,  // (B,A,H,W,4)
    const float* __restrict__ bboxes,   // (B,N,5)
    float* __restrict__ out)            // (B, P, N)
{
    __shared__ float4 sProp[TILE_P];    // decoded proposals (x1,y1,x2,y2)
    __shared__ float  sBox[N_ * 5];     // raw bbox rows, stride 5 (bank-conflict free)

    const int tid = threadIdx.x;
    const int b   = blockIdx.y;
    const int p0  = blockIdx.x * TILE_P;

    // ---- stage 64 bboxes (5 floats each = 80 float4 chunks) via async DMA ----
    if (tid < (N_ * 5 / 4)) {
        async_copy_b128(bboxes + (size_t)b * (N_ * 5) + tid * 4, &sBox[tid * 4]);
    }

    // ---- decode this block's 64 proposals while the async copy is in flight ----
    if (tid < TILE_P) {
        int p   = p0 + tid;
        int a   = p / (H_ * W_);
        int rem = p % (H_ * W_);
        int h   = rem / W_;
        int w   = rem % W_;
        float2 g   = ((const float2*)grid)[(b * H_ + h) * W_ + w];
        float2 an  = ((const float2*)anc)[a];
        float4 off = ((const float4*)offsets)[(size_t)b * P_ + p];
        float xc = g.x + off.x;
        float yc = g.y + off.y;
        float wp = an.x * __expf(off.z);
        float hp = an.y * __expf(off.w);
        sProp[tid] = make_float4(xc - 0.5f * wp, yc - 0.5f * hp,
                                 xc + 0.5f * wp, yc + 0.5f * hp);
    }

    wait_async0();
    __syncthreads();

    const int lane = tid & 31;
    const int wv   = tid >> 5;    // wave -> 16-proposal subtile
    const int hi   = lane >> 4;   // half-wave selector (M = r + 8*hi)
    const int lj   = lane & 15;   // N within 16-wide tile

    // A matrix (16x4 f32): col0 = proposal area, col1 = 1, cols 2/3 = 0.
    // Layout: lanes 0-15 hold M=lane, VGPR0=K0, VGPR1=K1; lanes 16-31 hold K2/K3 (zero).
    float a0 = 0.0f, a1 = 0.0f;
    if (lane < 16) {
        float4 pr = sProp[wv * 16 + lane];
        a0 = (pr.z - pr.x) * (pr.w - pr.y);
        a1 = 1.0f;
    }
    v2f Am = { a0, a1 };

    float* outbase = out + ((size_t)b * P_ + p0) * N_;

    #pragma unroll
    for (int t = 0; t < 4; ++t) {
        const int n = t * 16 + lj;
        const float bx1 = sBox[n * 5 + 0];
        const float by1 = sBox[n * 5 + 1];
        const float bx2 = sBox[n * 5 + 2];
        const float by2 = sBox[n * 5 + 3];

        // B matrix (4x16 f32): row0 = 1, row1 = box area, rows 2/3 = 0.
        // VGPR0: lanes 0-15 = row K0, lanes 16-31 = row K1; VGPR1 = rows K2/K3.
        float b0 = (lane < 16) ? 1.0f : (bx2 - bx1) * (by2 - by1);
        v2f Bm = { b0, 0.0f };

        v8f c = {};  // D[m][n] = aop_m + aob_n  (outer sum via rank-2 matmul)
        c = __builtin_amdgcn_wmma_f32_16x16x4_f32(
                /*neg_a=*/false, Am, /*neg_b=*/false, Bm,
                /*c_mod=*/(short)0, c, /*reuse_a=*/false, /*reuse_b=*/false);

        #pragma unroll
        for (int r = 0; r < 8; ++r) {
            const int pm = wv * 16 + r + 8 * hi;    // proposal row in block tile
            float4 pr = sProp[pm];                  // half-wave broadcast read
            float tlx = fmaxf(pr.x, bx1);
            float tly = fmaxf(pr.y, by1);
            float brx = fminf(pr.z, bx2);
            float bry = fminf(pr.w, by2);
            float iw = brx - tlx;
            float ih = bry - tly;
            float inter = ((tlx > brx) || (tly > bry)) ? 0.0f : iw * ih;
            float denom = c[r] - inter;             // aop + aob - aoi (from WMMA)
            outbase[(size_t)pm * N_ + n] = inter * __builtin_amdgcn_rcpf(denom);
        }
    }
}

extern "C" void kernel_launch(void* const* d_in, const int* in_sizes, int n_in,
                              void* d_out, int out_size, void* d_ws, size_t ws_size,
                              hipStream_t stream) {
    const float* anc     = (const float*)d_in[0];
    const float* grid    = (const float*)d_in[1];
    const float* offsets = (const float*)d_in[2];
    const float* bboxes  = (const float*)d_in[3];
    float* out = (float*)d_out;

    dim3 g(P_ / TILE_P, B_, 1);   // (441, 16)
    iou_fused_kernel<<<g, 128, 0, stream>>>(anc, grid, offsets, bboxes, out);
}